// soft_dice_cldice_36112085024849
// MI455X (gfx1250) — compile-verified
//
#include <hip/hip_runtime.h>
#include <math.h>

#define HH    512
#define WW    512
#define NIMG  16
#define NPL   32              // 16 pred planes + 16 true planes
#define PLANE (HH * WW)
#define NTOT  (NIMG * PLANE)

#define TW    64
#define TH    16
#define BLK   256
#define NBRED 512
#define NITER 20

// ---- CDNA5 feature detection (device pass only; host pass takes fallbacks) ----
#if defined(__gfx1250__)
#  ifdef __has_builtin
#    if __has_builtin(__builtin_amdgcn_wmma_f32_16x16x4_f32)
#      define HAVE_WMMA_F32X4 1
#    endif
#    if __has_builtin(__builtin_amdgcn_global_load_async_to_lds_b32)
#      define HAVE_ASYNC_LDS 1
#    endif
#    if __has_builtin(__builtin_amdgcn_s_wait_asynccnt)
#      define HAVE_WAIT_ASYNC 1
#    endif
#  endif
#endif

typedef __attribute__((ext_vector_type(2))) float v2f;
typedef __attribute__((ext_vector_type(8))) float v8f;

__device__ __forceinline__ float gelu_exact(float x) {
  // exact erf formulation (matches jax.nn.gelu approximate=False)
  return 0.5f * x * (1.0f + erff(x * 0.70710678118654752440f));
}

// 32-lane wave sum. On gfx1250 uses V_WMMA_F32_16X16X4_F32 with B = ones:
// A[m][0]=v_m (lanes 0-15), A[m][2]=v_{m+16} (lanes 16-31), others 0
// => D[m][n] = v_m + v_{m+16}; each lane sums its 8 C VGPRs (its half of M),
// then one xor-16 shuffle combines halves. EXEC must be all ones (callers
// invoke from fully converged 256-thread blocks).
__device__ __forceinline__ float wave_sum32(float v) {
#ifdef HAVE_WMMA_F32X4
  v2f a; a.x = v;    a.y = 0.0f;
  v2f b; b.x = 1.0f; b.y = 1.0f;
  v8f c = 0.0f;
  c = __builtin_amdgcn_wmma_f32_16x16x4_f32(false, a, false, b, (short)0, c,
                                            false, false);
  float hs = ((c[0] + c[1]) + (c[2] + c[3])) + ((c[4] + c[5]) + (c[6] + c[7]));
  hs += __shfl_xor(hs, 16, 32);
  return hs;
#else
  #pragma unroll
  for (int off = 16; off > 0; off >>= 1) v += __shfl_xor(v, off, 32);
  return v;
#endif
}

// Reduce 7 per-thread accumulators across a 256-thread block; thread 0 ends
// with totals in a[0..6].
__device__ __forceinline__ void block_reduce7(float* a) {
  __shared__ float redS[7][8];
  const int tid = threadIdx.x, wave = tid >> 5, lane = tid & 31;
  #pragma unroll
  for (int q = 0; q < 7; ++q) {
    float s = wave_sum32(a[q]);
    if (lane == 0) redS[q][wave] = s;
  }
  __syncthreads();
  if (tid == 0) {
    #pragma unroll
    for (int q = 0; q < 7; ++q) {
      float s = 0.0f;
      #pragma unroll
      for (int w = 0; w < 8; ++w) s += redS[q][w];
      a[q] = s;
    }
  }
}

// One fused soft-skel iteration:
//   E = erode(img) = min( vmin3(img), hmin3(img) )   (OOB contributes +inf)
//   D = dilate3x3(E)                                  (OOB contributes -inf)
//   delta = gelu(img - D)
//   skel  = first ? delta : skel + gelu(delta - skel*delta)
//   imgOut = E
// LDS tiling: img tile with +2 halo, E tile with +1 halo. Boundary semantics
// exactly match lax.reduce_window padding with -inf init.
__global__ __launch_bounds__(BLK) void cl_iter_kernel(
    const float* __restrict__ srcP, const float* __restrict__ srcT, int split,
    float* __restrict__ imgOut, const float* __restrict__ skelIn,
    float* __restrict__ skelOut, int first) {
  __shared__ float imgS[(TH + 4) * (TW + 4)];
  __shared__ float eS[(TH + 2) * (TW + 2)];

  const int p   = blockIdx.z;
  const int by  = blockIdx.y * TH;
  const int bx  = blockIdx.x * TW;
  const int tid = threadIdx.x;

  const float* src;
  size_t ibase;
  if (split) {  // first pass reads the original inputs (planes 0-15 = pred)
    src = (p < NIMG) ? srcP : srcT;
    ibase = (size_t)(p & (NIMG - 1)) * PLANE;
  } else {
    src = srcP;
    ibase = (size_t)p * PLANE;
  }
  const size_t obase = (size_t)p * PLANE;

  // ---- stage 1: img tile with +2 halo into LDS; out-of-image -> +inf ----
  for (int idx = tid; idx < (TH + 4) * (TW + 4); idx += BLK) {
    int ly = idx / (TW + 4), lx = idx - ly * (TW + 4);
    int gy = by + ly - 2, gx = bx + lx - 2;
    bool in = (gy >= 0) & (gy < HH) & (gx >= 0) & (gx < WW);
#ifdef HAVE_ASYNC_LDS
    if (in) {
      __builtin_amdgcn_global_load_async_to_lds_b32(
          (__attribute__((address_space(1))) int*)(src + ibase +
                                                   (size_t)gy * WW + gx),
          (__attribute__((address_space(3))) int*)&imgS[idx], 0, 0);
    } else {
      imgS[idx] = __builtin_inff();
    }
#else
    imgS[idx] = in ? src[ibase + (size_t)gy * WW + gx] : __builtin_inff();
#endif
  }
#ifdef HAVE_ASYNC_LDS
#  ifdef HAVE_WAIT_ASYNC
  __builtin_amdgcn_s_wait_asynccnt(0);
#  else
  asm volatile("s_wait_asynccnt 0" ::: "memory");
#  endif
#endif
  __syncthreads();

  // ---- stage 2: erode with +1 halo; positions outside image -> -inf ----
  for (int idx = tid; idx < (TH + 2) * (TW + 2); idx += BLK) {
    int ly = idx / (TW + 2), lx = idx - ly * (TW + 2);
    int gy = by + ly - 1, gx = bx + lx - 1;
    float e;
    if ((gy >= 0) & (gy < HH) & (gx >= 0) & (gx < WW)) {
      const float* c = &imgS[(ly + 1) * (TW + 4) + (lx + 1)];
      float vm = fminf(fminf(c[-(TW + 4)], c[0]), c[TW + 4]);
      float hm = fminf(fminf(c[-1], c[0]), c[1]);
      e = fminf(vm, hm);
    } else {
      e = -__builtin_inff();
    }
    eS[idx] = e;
  }
  __syncthreads();

  // ---- stage 3: dilate, delta, skel update, write eroded img ----
  for (int idx = tid; idx < TH * TW; idx += BLK) {
    int ly = idx / TW, lx = idx - ly * TW;
    const float* e0 = &eS[ly * (TW + 2) + lx];
    const float* e1 = e0 + (TW + 2);
    const float* e2 = e1 + (TW + 2);
    float d = fmaxf(fmaxf(e0[0], e0[1]), e0[2]);
    d = fmaxf(d, fmaxf(fmaxf(e1[0], e1[1]), e1[2]));
    d = fmaxf(d, fmaxf(fmaxf(e2[0], e2[1]), e2[2]));

    float img   = imgS[(ly + 2) * (TW + 4) + (lx + 2)];
    float delta = gelu_exact(img - d);

    const size_t o = obase + (size_t)(by + ly) * WW + (bx + lx);
    float s;
    if (first) {
      s = delta;  // skel = gelu(img - open(img))
    } else {
      float sk = skelIn[o];
      s = sk + gelu_exact(delta - sk * delta);
    }
    skelOut[o] = s;
    imgOut[o]  = e1[1];  // erode(img) at center -> next iteration's img
  }
}

// Stage-1 reduction: per-block partials of the 7 global sums.
__global__ __launch_bounds__(BLK) void cl_reduce_partial(
    const float* __restrict__ yt, const float* __restrict__ yp,
    const float* __restrict__ skel, float* __restrict__ partials) {
  float a[7] = {0.f, 0.f, 0.f, 0.f, 0.f, 0.f, 0.f};
  for (int i = blockIdx.x * BLK + threadIdx.x; i < NTOT;
       i += gridDim.x * BLK) {
    float t  = yt[i];
    float pr = yp[i];
    float sp = skel[i];         // planes 0-15: skel(pred)
    float st = skel[NTOT + i];  // planes 16-31: skel(true)
    a[0] += t;
    a[1] += pr;
    a[2] += t * pr;
    a[3] += sp;
    a[4] += sp * t;
    a[5] += st;
    a[6] += st * pr;
  }
  block_reduce7(a);
  if (threadIdx.x == 0) {
    #pragma unroll
    for (int q = 0; q < 7; ++q) partials[blockIdx.x * 8 + q] = a[q];
  }
}

// Stage-2 reduction + final dice/clDice formula (single block, deterministic).
__global__ __launch_bounds__(BLK) void cl_reduce_final(
    const float* __restrict__ partials, float* __restrict__ out, int nb) {
  float a[7] = {0.f, 0.f, 0.f, 0.f, 0.f, 0.f, 0.f};
  for (int b = threadIdx.x; b < nb; b += BLK) {
    #pragma unroll
    for (int q = 0; q < 7; ++q) a[q] += partials[b * 8 + q];
  }
  block_reduce7(a);
  if (threadIdx.x == 0) {
    float s_yt = a[0], s_yp = a[1], s_int = a[2];
    float s_sp = a[3], s_spyt = a[4], s_st = a[5], s_styp = a[6];
    float dice  = 1.0f - (2.0f * s_int + 1.0f) / (s_yt + s_yp + 1.0f);
    float tprec = (s_spyt + 1.0f) / (s_sp + 1.0f);
    float tsens = (s_styp + 1.0f) / (s_st + 1.0f);
    float cld   = 1.0f - 2.0f * (tprec * tsens) / (tprec + tsens);
    out[0] = 0.7f * dice + 0.3f * cld;  // (1-ALPHA)*dice + ALPHA*clDice
  }
}

extern "C" void kernel_launch(void* const* d_in, const int* in_sizes, int n_in,
                              void* d_out, int out_size, void* d_ws,
                              size_t ws_size, hipStream_t stream) {
  (void)in_sizes; (void)n_in; (void)out_size; (void)ws_size;
  const float* y_true = (const float*)d_in[0];
  const float* y_pred = (const float*)d_in[1];

  float* ws       = (float*)d_ws;
  float* imgA     = ws;                               // 32 planes
  float* imgB     = ws + (size_t)NPL * PLANE;         // 32 planes
  float* skel     = ws + 2 * (size_t)NPL * PLANE;     // 32 planes (in-place)
  float* partials = ws + 3 * (size_t)NPL * PLANE;     // NBRED*8 floats

  dim3 grid(WW / TW, HH / TH, NPL);

  // k = 0: reads original inputs, writes skel = gelu(img - open(img)),
  // imgA = erode(img).
  cl_iter_kernel<<<grid, BLK, 0, stream>>>(y_pred, y_true, 1, imgA, skel, skel,
                                           1);
  // k = 1..20: img = erode(img); skel += gelu(delta - skel*delta)
  float* cur = imgA;
  float* nxt = imgB;
  for (int k = 1; k <= NITER; ++k) {
    cl_iter_kernel<<<grid, BLK, 0, stream>>>(cur, nullptr, 0, nxt, skel, skel,
                                             0);
    float* t = cur; cur = nxt; nxt = t;
  }

  cl_reduce_partial<<<NBRED, BLK, 0, stream>>>(y_true, y_pred, skel, partials);
  cl_reduce_final<<<1, BLK, 0, stream>>>(partials, (float*)d_out, NBRED);
}